// GCN_3298534884317
// MI455X (gfx1250) — compile-verified
//
#include <hip/hip_runtime.h>

#define N_NODES_C 50000
#define FDIM      64
#define N_EDGES_C 800000

typedef float v2f __attribute__((ext_vector_type(2)));
typedef float v8f __attribute__((ext_vector_type(8)));

// ---------------------------------------------------------------------------
// Kernel 1: zero the workspace accumulators (agg[N*64] ++ cnt[N])
// ---------------------------------------------------------------------------
__global__ void gcn_init_ws(float* __restrict__ ws, int n) {
  int i = blockIdx.x * blockDim.x + threadIdx.x;
  if (i < n) ws[i] = 0.0f;
}

// ---------------------------------------------------------------------------
// Kernel 2: edge scatter. 16 threads per edge, float4 per thread.
// agg[dst] += feature[src]; cnt[dst] += 1. Atomics hit the L2-resident 12.8MB
// accumulator; the 205MB of gathered feature rows is the bandwidth cost.
// ---------------------------------------------------------------------------
__global__ void gcn_edge_scatter(const float* __restrict__ feat,
                                 const int* __restrict__ esrc,
                                 const int* __restrict__ edst,
                                 float* __restrict__ agg,
                                 float* __restrict__ cnt) {
  long t = (long)blockIdx.x * blockDim.x + threadIdx.x;
  long e = t >> 4;                 // 16 threads / edge
  if (e >= N_EDGES_C) return;
  int j = (int)(t & 15) * 4;       // 4 consecutive floats
  int s = esrc[e];
  int d = edst[e];
  const float4 v = *(const float4*)(feat + (long)s * FDIM + j);
  float* a = agg + (long)d * FDIM + j;
  atomicAdd(a + 0, v.x);
  atomicAdd(a + 1, v.y);
  atomicAdd(a + 2, v.z);
  atomicAdd(a + 3, v.w);
  if ((t & 15) == 0) atomicAdd(cnt + d, 1.0f);
}

// ---------------------------------------------------------------------------
// Kernel 3: fused mean-normalize + (h @ W^T + b) + ReLU via exact-fp32 WMMA.
// 128 threads = 4 wave32, 64 nodes per block. Each wave owns a 16-node strip
// and computes 4 16x16 output tiles with 16 chained V_WMMA_F32_16X16X4_F32.
//
// f32 16x16x4 fragment layout (ISA 7.12.2):
//   A (16x4, MxK): lane = M%16 striped twice; lanes 0-15 hold K={0,1} in
//                  v0,v1; lanes 16-31 hold K={2,3}.  -> float2 per lane.
//   B (4x16, KxN): symmetric with N = lane%16 (B loaded column-major: the
//                  lane holds W[n][k..k+1]).
//   C/D: VGPR r -> M=r (lanes 0-15) / M=r+8 (lanes 16-31), N = lane%16.
// ---------------------------------------------------------------------------
#define TILE_NODES 64
#define LDSPITCH   68   // 64 + 4 pad: breaks 64-bank row aliasing, keeps 8B align

__global__ __launch_bounds__(128) void gcn_apply(
    const float* __restrict__ feat,
    const float* __restrict__ agg,
    const float* __restrict__ cnt,
    const float* __restrict__ Wm,    // [64][64] row = out-feature n, col = k
    const float* __restrict__ bias,  // [64]
    float* __restrict__ out) {
  __shared__ float sW[FDIM][LDSPITCH];
  __shared__ float sH[TILE_NODES][LDSPITCH];
  __shared__ float sB[FDIM];

  const int tid  = threadIdx.x;      // 0..127
  const int wave = tid >> 5;         // 0..3
  const int lane = tid & 31;
  const int baseNode = blockIdx.x * TILE_NODES;
  // Block-uniform: every row of this tile is in range (true for 781/782 blocks)
  const bool fullTile = (baseNode + TILE_NODES) <= N_NODES_C;

  // Stage W (16KB) and bias into LDS.
  for (int i = tid; i < FDIM * FDIM; i += 128)
    sW[i >> 6][i & 63] = Wm[i];
  if (tid < FDIM) sB[tid] = bias[tid];

  // Stage normalized h tile: mean if cnt>0 else original feature.
  for (int i = tid; i < TILE_NODES * FDIM; i += 128) {
    int r = i >> 6;
    int c = i & 63;
    int node = baseNode + r;
    float v = 0.0f;
    if (fullTile || node < N_NODES_C) {
      float cn = cnt[node];
      v = (cn > 0.0f) ? agg[(long)node * FDIM + c] / cn
                      : feat[(long)node * FDIM + c];
    }
    sH[r][c] = v;
  }
  __syncthreads();

  const int mlocal = lane & 15;          // A row / B col within tile
  const int khalf  = (lane >> 4) << 1;   // 0 for lanes 0-15, 2 for 16-31
  const int rowA   = wave * 16 + mlocal; // local node row in sH

  for (int nt = 0; nt < 4; ++nt) {
    v8f acc = {};
    const int rowB = nt * 16 + mlocal;   // out-feature row of W
#pragma unroll
    for (int kk = 0; kk < FDIM; kk += 4) {
      v2f a = *(const v2f*)&sH[rowA][kk + khalf];
      v2f b = *(const v2f*)&sW[rowB][kk + khalf];
      acc = __builtin_amdgcn_wmma_f32_16x16x4_f32(
          /*neg_a=*/false, a, /*neg_b=*/false, b,
          /*c_mod=*/(short)0, acc, /*reuse_a=*/false, /*reuse_b=*/false);
    }
    // Epilogue: bias + ReLU + store per documented C/D layout.
    const int mbase = baseNode + wave * 16 + ((lane >> 4) << 3);
    const int ncol  = nt * 16 + mlocal;
    const float bv  = sB[ncol];
    if (fullTile) {
      // Fast path: straight-line unconditional stores (no exec-mask churn).
#pragma unroll
      for (int r = 0; r < 8; ++r) {
        float v = acc[r] + bv;
        out[(long)(mbase + r) * FDIM + ncol] = v > 0.0f ? v : 0.0f;
      }
    } else {
#pragma unroll
      for (int r = 0; r < 8; ++r) {
        int m = mbase + r;
        if (m < N_NODES_C) {
          float v = acc[r] + bv;
          out[(long)m * FDIM + ncol] = v > 0.0f ? v : 0.0f;
        }
      }
    }
  }
}

// ---------------------------------------------------------------------------
extern "C" void kernel_launch(void* const* d_in, const int* in_sizes, int n_in,
                              void* d_out, int out_size, void* d_ws, size_t ws_size,
                              hipStream_t stream) {
  const float* feat = (const float*)d_in[0];
  const int*   esrc = (const int*)d_in[1];
  const int*   edst = (const int*)d_in[2];
  const float* Wm   = (const float*)d_in[3];
  const float* bias = (const float*)d_in[4];
  float* out = (float*)d_out;

  float* agg = (float*)d_ws;                       // [N_NODES * 64]
  float* cnt = agg + (size_t)N_NODES_C * FDIM;     // [N_NODES]

  // 1) zero accumulators
  int zn = N_NODES_C * (FDIM + 1);
  gcn_init_ws<<<(zn + 255) / 256, 256, 0, stream>>>(agg, zn);

  // 2) edge scatter (16 threads per edge)
  long tE = (long)N_EDGES_C * 16;
  gcn_edge_scatter<<<(unsigned)((tE + 255) / 256), 256, 0, stream>>>(
      feat, esrc, edst, agg, cnt);

  // 3) fused normalize + GEMM + bias + ReLU
  gcn_apply<<<(N_NODES_C + TILE_NODES - 1) / TILE_NODES, 128, 0, stream>>>(
      feat, agg, cnt, Wm, bias, out);
}